// CrossEntropyLoss_86629490360769
// MI455X (gfx1250) — compile-verified
//
#include <hip/hip_runtime.h>
#include <stdint.h>

// Cross-entropy loss: predicts [4096, 32000] f32, targets [4096] int.
// out = mean_i( logsumexp(predicts[i,:]) - predicts[i, targets[i]] )
//
// 524 MB @ 23.3 TB/s ~= 22.5 us floor; ~131M v_exp_f32 is near-parity with
// that, so minimize trans ops: stage 16 elements in registers per online
// merge (17 exps / 16 elems). Stream each 128 KB row into LDS with CDNA5
// async global->LDS DMA, consumed in a single pass pipelined under the
// stream via staged s_wait_asynccnt (per-wave in-order completion; each
// thread reads exactly the float4s it issued).

#define CE_BATCH   4096
#define CE_NCLS    32000
#define CE_NV4     (CE_NCLS / 4)          // 8000 float4 per row
#define CE_TPB     512                     // 16 waves; 2 blocks/WGP (LDS-limited)
#define CE_ITERS   16                      // ceil(8000/512) async loads per lane
#define CE_LN2     0.69314718055994530942f
#define CE_LOG2E   1.44269504088896340736f

__device__ __forceinline__ float ce_max4(const float4 x) {
    return fmaxf(fmaxf(x.x, x.y), fmaxf(x.z, x.w));
}

__device__ __forceinline__ float ce_expsum4(const float4 x, const float nmK) {
    const float e0 = __builtin_amdgcn_exp2f(__builtin_fmaf(x.x, CE_LOG2E, nmK));
    const float e1 = __builtin_amdgcn_exp2f(__builtin_fmaf(x.y, CE_LOG2E, nmK));
    const float e2 = __builtin_amdgcn_exp2f(__builtin_fmaf(x.z, CE_LOG2E, nmK));
    const float e3 = __builtin_amdgcn_exp2f(__builtin_fmaf(x.w, CE_LOG2E, nmK));
    return (e0 + e1) + (e2 + e3);
}

// Process 4 chunks (16 elements) with one online merge: 17 exps / 16 elems.
template <bool TAIL>
__device__ __forceinline__ void ce_group(const float4* __restrict__ rowv,
                                         const int base, const int lane,
                                         float& m, float& s)
{
    const float4 x0 = rowv[base + 0 * CE_TPB + lane];
    const float4 x1 = rowv[base + 1 * CE_TPB + lane];
    const float4 x2 = rowv[base + 2 * CE_TPB + lane];
    float4 x3;
    if (TAIL) {
        const int v = base + 3 * CE_TPB + lane;
        x3 = rowv[v < CE_NV4 ? v : (CE_NV4 - 1)];
        if (v >= CE_NV4)   // exp -> 0, max-neutral
            x3 = make_float4(-INFINITY, -INFINITY, -INFINITY, -INFINITY);
    } else {
        x3 = rowv[base + 3 * CE_TPB + lane];
    }

    const float m16 = fmaxf(fmaxf(ce_max4(x0), ce_max4(x1)),
                            fmaxf(ce_max4(x2), ce_max4(x3)));
    const float nm  = fmaxf(m, m16);
    const float nmK = -nm * CE_LOG2E;      // exp arg: fma(x, log2e, nmK)

    const float t01 = ce_expsum4(x0, nmK) + ce_expsum4(x1, nmK);
    const float t23 = ce_expsum4(x2, nmK) + ce_expsum4(x3, nmK);
    s = s * __builtin_amdgcn_exp2f(__builtin_fmaf(m, CE_LOG2E, nmK))
        + (t01 + t23);
    m = nm;
}

__global__ __launch_bounds__(CE_TPB)
void ce_row_kernel(const float* __restrict__ predicts,
                   const int*   __restrict__ targets,
                   float*       __restrict__ row_loss)
{
    __shared__ float row[CE_NCLS];   // 128000 B: whole row resident in LDS
    __shared__ float smax[CE_TPB];
    __shared__ float ssum[CE_TPB];

    const int row_id = blockIdx.x;
    const int lane   = threadIdx.x;

    // ---- Issue all 16 async b128 loads (ASYNCcnt max 63) ----
    {
        const uint64_t sbase =
            (uint64_t)(uintptr_t)(predicts + (size_t)row_id * CE_NCLS);
        const uint32_t lds_base = (uint32_t)(uintptr_t)(&row[0]);
        #pragma unroll
        for (int i = 0; i < CE_ITERS; ++i) {
            int v = i * CE_TPB + lane;
            if (v > CE_NV4 - 1) v = CE_NV4 - 1;   // tail: duplicate-clamp
            const uint32_t voff = (uint32_t)v * 16u;
            const uint32_t ldsa = lds_base + voff;
            asm volatile(
                "global_load_async_to_lds_b128 %0, %1, %2"
                :
                : "v"(ldsa), "v"(voff), "s"(sbase)
                : "memory");
        }
    }

    // ---- Single online pass, pipelined under the DMA stream ----
    float m = -INFINITY;
    float s = 0.0f;
    const float4* rowv = reinterpret_cast<const float4*>(row);

    asm volatile("s_wait_asynccnt 12" ::: "memory");   // chunks 0..3 ready
    ce_group<false>(rowv, 0 * CE_TPB, lane, m, s);
    asm volatile("s_wait_asynccnt 8"  ::: "memory");   // chunks 4..7 ready
    ce_group<false>(rowv, 4 * CE_TPB, lane, m, s);
    asm volatile("s_wait_asynccnt 4"  ::: "memory");   // chunks 8..11 ready
    ce_group<false>(rowv, 8 * CE_TPB, lane, m, s);
    asm volatile("s_wait_asynccnt 0"  ::: "memory");   // all chunks ready
    ce_group<true >(rowv, 12 * CE_TPB, lane, m, s);

    // ---- Cross-thread merge of (m, s) pairs with rescale ----
    smax[lane] = m;
    ssum[lane] = s;
    __syncthreads();
    #pragma unroll
    for (int k = CE_TPB / 2; k > 0; k >>= 1) {
        if (lane < k) {
            const float ma = smax[lane], mb = smax[lane + k];
            const float sa = ssum[lane], sb = ssum[lane + k];
            const float nm = fmaxf(ma, mb);
            smax[lane] = nm;
            ssum[lane] = sa * __builtin_amdgcn_exp2f((ma - nm) * CE_LOG2E)
                       + sb * __builtin_amdgcn_exp2f((mb - nm) * CE_LOG2E);
        }
        __syncthreads();
    }

    if (lane == 0) {
        // All waves passed asynccnt==0 before the barriers: row fully valid.
        const int   t   = targets[row_id];
        const float xt  = row[t];
        const float lse = smax[0] + __builtin_amdgcn_logf(ssum[0]) * CE_LN2;
        row_loss[row_id] = lse - xt;
    }
}

// Deterministic final mean over 4096 per-row losses (no float atomics).
__global__ __launch_bounds__(256)
void ce_mean_kernel(const float* __restrict__ row_loss,
                    float*       __restrict__ out)
{
    __shared__ float red[256];
    float s = 0.0f;
    for (int i = threadIdx.x; i < CE_BATCH; i += 256) s += row_loss[i];
    red[threadIdx.x] = s;
    __syncthreads();
    #pragma unroll
    for (int k = 128; k > 0; k >>= 1) {
        if (threadIdx.x < k) red[threadIdx.x] += red[threadIdx.x + k];
        __syncthreads();
    }
    if (threadIdx.x == 0) out[0] = red[0] * (1.0f / (float)CE_BATCH);
}

extern "C" void kernel_launch(void* const* d_in, const int* in_sizes, int n_in,
                              void* d_out, int out_size, void* d_ws, size_t ws_size,
                              hipStream_t stream) {
    (void)in_sizes; (void)n_in; (void)out_size; (void)ws_size;
    const float* predicts = (const float*)d_in[0];
    const int*   targets  = (const int*)d_in[1];
    float*       out      = (float*)d_out;
    float*       row_loss = (float*)d_ws;   // 4096 floats of scratch

    ce_row_kernel<<<CE_BATCH, CE_TPB, 0, stream>>>(predicts, targets, row_loss);
    ce_mean_kernel<<<1, 256, 0, stream>>>(row_loss, out);
}